// SCCPowerIteration_19550691132071
// MI455X (gfx1250) — compile-verified
//
#include <hip/hip_runtime.h>
#include <hip/hip_bf16.h>

// ---------------------------------------------------------------------------
// SCC power iteration on MI455X (gfx1250, wave32).
// matrix = adj^2 (HBM streamed once, then L2-resident: 64MB < 192MB L2).
// Matvecs M@v and M^T@vt done with V_WMMA_F32_16X16X4_F32 (full fp32):
//   rows kernel: A = M tile (16x4), B = broadcast(v)   -> y rows
//   cols kernel: A = broadcast(vt), B = M tile (4x16)  -> yt cols (no transpose!)
// K-split 16 ways (4096 waves/matvec), 4x unrolled WMMA loops, and all
// dimensions compile-time so loads fold into base+immediate-offset addressing.
// ---------------------------------------------------------------------------

typedef __attribute__((ext_vector_type(2))) float v2f;
typedef __attribute__((ext_vector_type(8))) float v8f;

#define D_DIM 4096
#define EPS 1e-6f
#define DIAG_BOOST 100.0f
#define KSPLIT 16
#define KCHUNK (D_DIM / KSPLIT)        // 256 columns/rows per wave sweep
#define WAVES_PER_BLOCK 8

// -------------------------- matrix = adj * adj ------------------------------
__global__ void square_kernel(const float* __restrict__ adj,
                              float* __restrict__ mat) {
    int t = blockIdx.x * blockDim.x + threadIdx.x;
    float4 a = reinterpret_cast<const float4*>(adj)[t];
    float4 r;
    r.x = a.x * a.x; r.y = a.y * a.y; r.z = a.z * a.z; r.w = a.w * a.w;
    reinterpret_cast<float4*>(mat)[t] = r;
}

// -------------------------- v = vt = 1/sqrt(D) ------------------------------
__global__ void init_kernel(float* __restrict__ v, float* __restrict__ vt) {
    int t = blockIdx.x * blockDim.x + threadIdx.x;
    v[t] = 0.015625f;   // 1/sqrt(4096)
    vt[t] = 0.015625f;
}

// -------------------- s[0]=sum(v), s[1]=sum(vt) -----------------------------
__global__ void sum_kernel(const float* __restrict__ v,
                           const float* __restrict__ vt,
                           float* __restrict__ s) {
    const float* src = (blockIdx.x == 0) ? v : vt;
    __shared__ float red[256];
    float acc = 0.0f;
    #pragma unroll
    for (int j = 0; j < D_DIM / 256; ++j) acc += src[threadIdx.x + j * 256];
    red[threadIdx.x] = acc;
    __syncthreads();
    for (int off = 128; off > 0; off >>= 1) {
        if ((int)threadIdx.x < off) red[threadIdx.x] += red[threadIdx.x + off];
        __syncthreads();
    }
    if (threadIdx.x == 0) s[blockIdx.x] = red[0];
}

// ------------------- y += M[rows] @ v  (WMMA f32 16x16x4) -------------------
// One wave owns 16 rows, sweeps a K-slice of KCHUNK columns.
__global__ void matvec_rows_wmma(const float* __restrict__ M,
                                 const float* __restrict__ v,
                                 float* __restrict__ y) {
    const int lane = threadIdx.x & 31;
    const int wave = threadIdx.x >> 5;
    const int half = lane >> 4;     // 0: K={0,1}, 1: K={2,3}
    const int l16  = lane & 15;
    const int row0 = (blockIdx.x * WAVES_PER_BLOCK + wave) * 16;
    const int kb   = blockIdx.y * KCHUNK;

    const float* Mp = M + (size_t)(row0 + l16) * D_DIM + kb + 2 * half;
    const float* vp = v + kb + 2 * half;
    v8f acc = {};
    #pragma unroll 4
    for (int k = 0; k < KCHUNK; k += 4) {
        v2f a, b;
        a.x = Mp[k + 0];             // A: 16x4 tile of M, row-major
        a.y = Mp[k + 1];
        b.x = vp[k + 0];             // B[k][j] = v[k]  (all columns)
        b.y = vp[k + 1];
        acc = __builtin_amdgcn_wmma_f32_16x16x4_f32(false, a, false, b,
                                                    (short)0, acc, false, false);
    }
    // D rows replicate y across columns: lane 0 -> rows 0..7, lane 16 -> rows 8..15
    if (l16 == 0) {
        #pragma unroll
        for (int r = 0; r < 8; ++r)
            atomicAdd(&y[row0 + half * 8 + r], acc[r]);
    }
}

// ------------------- yt += M^T[cols] @ vt  (WMMA f32 16x16x4) ---------------
// One wave owns 16 columns, sweeps a K-slice of KCHUNK rows.
__global__ void matvec_cols_wmma(const float* __restrict__ M,
                                 const float* __restrict__ vt,
                                 float* __restrict__ yt) {
    const int lane = threadIdx.x & 31;
    const int wave = threadIdx.x >> 5;
    const int half = lane >> 4;
    const int l16  = lane & 15;
    const int col0 = (blockIdx.x * WAVES_PER_BLOCK + wave) * 16;
    const int kb   = blockIdx.y * KCHUNK;

    // base: M[kb + 2*half][col0 + l16]; consecutive rows are +D_DIM (16KB imm)
    const float* Mp  = M + (size_t)(kb + 2 * half) * D_DIM + col0 + l16;
    const float* vtp = vt + kb + 2 * half;
    v8f acc = {};
    #pragma unroll 4
    for (int k = 0; k < KCHUNK; k += 4) {
        v2f a, b;
        a.x = vtp[k + 0];                            // A[i][k] = vt[k] (all rows)
        a.y = vtp[k + 1];
        b.x = Mp[(size_t)(k + 0) * D_DIM];           // B: 4x16 tile of M
        b.y = Mp[(size_t)(k + 1) * D_DIM];
        acc = __builtin_amdgcn_wmma_f32_16x16x4_f32(false, a, false, b,
                                                    (short)0, acc, false, false);
    }
    // D columns replicate yt across rows: lane l16 (half 0), component 0 = yt[col0+l16]
    if (half == 0) atomicAdd(&yt[col0 + l16], acc[0]);
}

// -------- v = (y + eps*s)/||y + eps*s||  (blockIdx 0: y->v, 1: yt->vt) ------
__global__ void normalize_kernel(const float* __restrict__ y,
                                 const float* __restrict__ yt,
                                 const float* __restrict__ s,
                                 float* __restrict__ v,
                                 float* __restrict__ vt) {
    const float* src = (blockIdx.x == 0) ? y : yt;
    float* dst       = (blockIdx.x == 0) ? v : vt;
    const float eps_s = EPS * s[blockIdx.x];
    float w[4];
    float acc = 0.0f;
    #pragma unroll
    for (int j = 0; j < 4; ++j) {
        int i = threadIdx.x + j * 1024;
        w[j] = src[i] + eps_s;
        acc += w[j] * w[j];
    }
    __shared__ float red[1024];
    red[threadIdx.x] = acc;
    __syncthreads();
    for (int off = 512; off > 0; off >>= 1) {
        if ((int)threadIdx.x < off) red[threadIdx.x] += red[threadIdx.x + off];
        __syncthreads();
    }
    const float inv = 1.0f / sqrtf(red[0]);
    #pragma unroll
    for (int j = 0; j < 4; ++j)
        dst[threadIdx.x + j * 1024] = w[j] * inv;
}

// -------------------------- s[2] = <vt, v> ----------------------------------
__global__ void dot_kernel(const float* __restrict__ v,
                           const float* __restrict__ vt,
                           float* __restrict__ s) {
    __shared__ float red[1024];
    float acc = 0.0f;
    #pragma unroll
    for (int j = 0; j < D_DIM / 1024; ++j) {
        int i = threadIdx.x + j * 1024;
        acc += v[i] * vt[i];
    }
    red[threadIdx.x] = acc;
    __syncthreads();
    for (int off = 512; off > 0; off >>= 1) {
        if ((int)threadIdx.x < off) red[threadIdx.x] += red[threadIdx.x + off];
        __syncthreads();
    }
    if (threadIdx.x == 0) s[0] = red[0];
}

// --------------- grad = outer(vt,v)/dot + 100*I  (float4 stores) ------------
__global__ void gradient_kernel(const float* __restrict__ v,
                                const float* __restrict__ vt,
                                const float* __restrict__ dot,
                                float* __restrict__ grad) {
    size_t t = (size_t)blockIdx.x * blockDim.x + threadIdx.x;
    size_t base = t * 4;
    int i  = (int)(base / D_DIM);
    int j0 = (int)(base % D_DIM);
    float a = vt[i] / dot[0];
    float4 r;
    r.x = a * v[j0 + 0] + ((j0 + 0 == i) ? DIAG_BOOST : 0.0f);
    r.y = a * v[j0 + 1] + ((j0 + 1 == i) ? DIAG_BOOST : 0.0f);
    r.z = a * v[j0 + 2] + ((j0 + 2 == i) ? DIAG_BOOST : 0.0f);
    r.w = a * v[j0 + 3] + ((j0 + 3 == i) ? DIAG_BOOST : 0.0f);
    reinterpret_cast<float4*>(grad)[t] = r;
}

// ---------------------------------------------------------------------------
extern "C" void kernel_launch(void* const* d_in, const int* in_sizes, int n_in,
                              void* d_out, int out_size, void* d_ws, size_t ws_size,
                              hipStream_t stream) {
    (void)in_sizes; (void)n_in; (void)out_size; (void)ws_size;
    const size_t dd = (size_t)D_DIM * D_DIM;

    const float* adj = (const float*)d_in[0];
    float* grad_out  = (float*)d_out;            // output 0: gradient (d*d)
    float* mat_out   = (float*)d_out + dd;       // output 1: matrix   (d*d)

    float* v  = (float*)d_ws;
    float* vt = v  + D_DIM;
    float* y  = vt + D_DIM;
    float* yt = y  + D_DIM;   // y and yt contiguous -> single memset
    float* s  = yt + D_DIM;   // s[0]=sum(v), s[1]=sum(vt), s[2]=<vt,v>

    // 1) matrix = adj^2 (streamed once; stays L2-resident for the iterations)
    square_kernel<<<(int)(dd / 4 / 256), 256, 0, stream>>>(adj, mat_out);

    // 2) v = vt = normalize(ones)
    init_kernel<<<D_DIM / 256, 256, 0, stream>>>(v, vt);

    // 3) 5 power iterations
    const dim3 mv_grid(D_DIM / (16 * WAVES_PER_BLOCK), KSPLIT);
    for (int it = 0; it < 5; ++it) {
        sum_kernel<<<2, 256, 0, stream>>>(v, vt, s);
        hipMemsetAsync(y, 0, 2 * (size_t)D_DIM * sizeof(float), stream);
        matvec_rows_wmma<<<mv_grid, 32 * WAVES_PER_BLOCK, 0, stream>>>(mat_out, v, y);
        matvec_cols_wmma<<<mv_grid, 32 * WAVES_PER_BLOCK, 0, stream>>>(mat_out, vt, yt);
        normalize_kernel<<<2, 1024, 0, stream>>>(y, yt, s, v, vt);
    }

    // 4) gradient = outer(vt, v)/<vt,v> + 100*I
    dot_kernel<<<1, 1024, 0, stream>>>(v, vt, s + 2);
    gradient_kernel<<<(int)(dd / 4 / 256), 256, 0, stream>>>(v, vt, s + 2, grad_out);
}